// MoeLayer_35845797053217
// MI455X (gfx1250) — compile-verified
//
#include <hip/hip_runtime.h>
#include <hip/hip_bf16.h>

typedef __attribute__((ext_vector_type(2))) float v2f;
typedef __attribute__((ext_vector_type(8))) float v8f;

#define T_TOKENS 8192
#define DK 2048          // D_IN
#define DN 2048          // D_OUT
#define NE 8             // sparse experts

#define MT 128           // block tile M
#define NT 128           // block tile N
#define KT 32            // K chunk

// ======================================================================
// Kernel 1: router — one wave32 per token.
// logits = x @ W_router + b_router ; softmax ; top-2 ; renormalize.
// ======================================================================
__global__ __launch_bounds__(256) void moe_router(
    const float* __restrict__ x, const float* __restrict__ Wr,
    const float* __restrict__ br, int* __restrict__ top_idx,
    float* __restrict__ top_w)
{
    const int wave = threadIdx.x >> 5;
    const int lane = threadIdx.x & 31;
    const int t = blockIdx.x * 8 + wave;

    float acc[NE];
#pragma unroll
    for (int e = 0; e < NE; ++e) acc[e] = 0.f;

    const float* xr = x + (size_t)t * DK;
    for (int d = lane; d < DK; d += 32) {
        const float xv = xr[d];                      // lanes: consecutive d -> coalesced
        const float4* w4 = (const float4*)(Wr + (size_t)d * NE);
        const float4 wa = w4[0], wb = w4[1];
        acc[0] += xv * wa.x; acc[1] += xv * wa.y;
        acc[2] += xv * wa.z; acc[3] += xv * wa.w;
        acc[4] += xv * wb.x; acc[5] += xv * wb.y;
        acc[6] += xv * wb.z; acc[7] += xv * wb.w;
    }
#pragma unroll
    for (int off = 16; off > 0; off >>= 1)
#pragma unroll
        for (int e = 0; e < NE; ++e) acc[e] += __shfl_xor(acc[e], off, 32);

    if (lane == 0) {
        float lg[NE], mx = -1e30f;
#pragma unroll
        for (int e = 0; e < NE; ++e) { lg[e] = acc[e] + br[e]; mx = fmaxf(mx, lg[e]); }
        float p[NE], se = 0.f;
#pragma unroll
        for (int e = 0; e < NE; ++e) { p[e] = __expf(lg[e] - mx); se += p[e]; }
        int i0 = 0; float v0 = p[0];
#pragma unroll
        for (int e = 1; e < NE; ++e) if (p[e] > v0) { v0 = p[e]; i0 = e; }
        int i1 = -1; float v1 = -1e30f;
#pragma unroll
        for (int e = 0; e < NE; ++e) if (e != i0 && p[e] > v1) { v1 = p[e]; i1 = e; }
        v0 /= se; v1 /= se;                           // softmax probs of top-2
        const float rs = 1.0f / (v0 + v1 + 1e-6f);    // renormalize (matches reference)
        top_idx[2 * t + 0] = i0; top_idx[2 * t + 1] = i1;
        top_w [2 * t + 0] = v0 * rs; top_w[2 * t + 1] = v1 * rs;
    }
}

// ======================================================================
// Kernel 2: deterministic stream compaction into per-(expert,slot) lists.
// 16 blocks: blockIdx.x = expert*2 + slot. Token order preserved.
// ======================================================================
__global__ __launch_bounds__(256) void moe_build_lists(
    const int* __restrict__ top_idx, const float* __restrict__ top_w,
    int* __restrict__ list_idx, float* __restrict__ list_w,
    int* __restrict__ counts)
{
    const int b = blockIdx.x;
    const int e = b >> 1, s = b & 1;
    const int tid = threadIdx.x;
    const int lane = tid & 31, w = tid >> 5;

    __shared__ int waveSums[8];
    __shared__ int baseOff;
    if (tid == 0) baseOff = 0;
    __syncthreads();

    const int lbase = b * T_TOKENS;
    for (int t0 = 0; t0 < T_TOKENS; t0 += 256) {
        const int t = t0 + tid;
        const int flag = (top_idx[2 * t + s] == e) ? 1 : 0;
        const unsigned m32 = (unsigned)__ballot(flag);
        const int lp = __popc(m32 & ((1u << lane) - 1u));
        if (lane == 0) waveSums[w] = __popc(m32);
        __syncthreads();
        int woff = 0;
#pragma unroll
        for (int j = 0; j < 8; ++j) if (j < w) woff += waveSums[j];
        const int pos = baseOff + woff + lp;
        if (flag) {
            list_idx[lbase + pos] = t;
            list_w [lbase + pos] = top_w[2 * t + s];
        }
        __syncthreads();
        if (tid == 0) {
            int tot = 0;
#pragma unroll
            for (int j = 0; j < 8; ++j) tot += waveSums[j];
            baseOff += tot;
        }
        __syncthreads();
    }
    if (tid == 0) counts[b] = baseOff;
}

// ======================================================================
// WMMA f32 GEMM core: block tile 128x128, 8 waves (4x2), wave tile 32x64.
// A in LDS row-major (padded); B in LDS k-paired layout Bs[k/2][n][2] so a
// B fragment (B[kb][n], B[kb+1][n]) is one aligned ds_load_b64 (kb even).
// 8 v_wmma_f32_16x16x4_f32 per k-step per wave, 6 LDS b64 loads.
// ======================================================================

// Kernel 3: y = x @ W_shared + b_shared + gates @ b_sparse  (full dense write)
__global__ __launch_bounds__(256) void moe_shared_gemm(
    const float* __restrict__ x, const float* __restrict__ Wsh,
    const float* __restrict__ bsh, const float* __restrict__ bsp,
    const int* __restrict__ top_idx, const float* __restrict__ top_w,
    float* __restrict__ y)
{
    __shared__ float As[MT][36];            // 18432 B, row stride 144 B
    __shared__ float Bs[KT / 2][NT + 2][2]; // 16640 B, b64-aligned pairs

    const int tid = threadIdx.x;
    const int lane = tid & 31, w = tid >> 5;
    const int wm = w >> 1, wn = w & 1;      // 4 x 2 wave grid
    const int hi = lane >> 4, ml = lane & 15;
    const int m0 = blockIdx.x * MT;
    const int n0 = blockIdx.y * NT;

    // A fill: row = tid/2, half-row (16 floats) per thread, 4x float4
    const int ar = tid >> 1, ac0 = (tid & 1) << 4;
    // B fill: column per thread, 8 k-pairs per thread
    const int bn = tid & 127, bh = (tid >> 7) << 3;   // pair base 0 or 8

    v8f acc[2][4] = {};

    for (int kk = 0; kk < DK; kk += KT) {
        // ---- stage A (tokens x K) ----
        const float* xa = x + (size_t)(m0 + ar) * DK + kk + ac0;
#pragma unroll
        for (int j = 0; j < 4; ++j)
            *(float4*)&As[ar][ac0 + 4 * j] = *(const float4*)(xa + 4 * j);
        // ---- stage B (K x N) into k-paired layout ----
#pragma unroll
        for (int i = 0; i < 8; ++i) {
            const int p = bh + i;                       // k-pair index
            const float b0 = Wsh[(size_t)(kk + 2 * p) * DN + n0 + bn];
            const float b1 = Wsh[(size_t)(kk + 2 * p + 1) * DN + n0 + bn];
            const v2f pv = { b0, b1 };
            *(v2f*)&Bs[p][bn][0] = pv;
        }
        if (kk + KT < DK)  // global_prefetch_b8 of next B chunk
            __builtin_prefetch(Wsh + (size_t)(kk + KT + 2 * bh) * DN + n0 + bn, 0, 0);
        __syncthreads();

#pragma unroll
        for (int ks = 0; ks < KT; ks += 4) {
            const int kb = ks + (hi << 1);              // even
            const int kp = kb >> 1;
            const v2f a0 = *(const v2f*)&As[wm * 32 + ml][kb];
            const v2f a1 = *(const v2f*)&As[wm * 32 + 16 + ml][kb];
            const v2f b0 = *(const v2f*)&Bs[kp][wn * 64 + ml][0];
            const v2f b1 = *(const v2f*)&Bs[kp][wn * 64 + 16 + ml][0];
            const v2f b2 = *(const v2f*)&Bs[kp][wn * 64 + 32 + ml][0];
            const v2f b3 = *(const v2f*)&Bs[kp][wn * 64 + 48 + ml][0];
            acc[0][0] = __builtin_amdgcn_wmma_f32_16x16x4_f32(false, a0, false, b0, (short)0, acc[0][0], false, false);
            acc[0][1] = __builtin_amdgcn_wmma_f32_16x16x4_f32(false, a0, false, b1, (short)0, acc[0][1], false, false);
            acc[0][2] = __builtin_amdgcn_wmma_f32_16x16x4_f32(false, a0, false, b2, (short)0, acc[0][2], false, false);
            acc[0][3] = __builtin_amdgcn_wmma_f32_16x16x4_f32(false, a0, false, b3, (short)0, acc[0][3], false, false);
            acc[1][0] = __builtin_amdgcn_wmma_f32_16x16x4_f32(false, a1, false, b0, (short)0, acc[1][0], false, false);
            acc[1][1] = __builtin_amdgcn_wmma_f32_16x16x4_f32(false, a1, false, b1, (short)0, acc[1][1], false, false);
            acc[1][2] = __builtin_amdgcn_wmma_f32_16x16x4_f32(false, a1, false, b2, (short)0, acc[1][2], false, false);
            acc[1][3] = __builtin_amdgcn_wmma_f32_16x16x4_f32(false, a1, false, b3, (short)0, acc[1][3], false, false);
        }
        __syncthreads();
    }

    // epilogue: + b_shared + w0*b_sparse[e0] + w1*b_sparse[e1]
    float bv[4];
#pragma unroll
    for (int sn = 0; sn < 4; ++sn) bv[sn] = bsh[n0 + wn * 64 + sn * 16 + ml];
#pragma unroll
    for (int sm = 0; sm < 2; ++sm)
#pragma unroll
        for (int v = 0; v < 8; ++v) {
            const int t = m0 + wm * 32 + sm * 16 + v + (hi << 3);
            const int e0 = top_idx[2 * t], e1 = top_idx[2 * t + 1];
            const float g0 = top_w[2 * t], g1 = top_w[2 * t + 1];
#pragma unroll
            for (int sn = 0; sn < 4; ++sn) {
                const int o = n0 + wn * 64 + sn * 16 + ml;
                y[(size_t)t * DN + o] = acc[sm][sn][v] + bv[sn]
                    + g0 * bsp[(size_t)e0 * DN + o] + g1 * bsp[(size_t)e1 * DN + o];
            }
        }
}

// Kernel 4: per-expert gathered GEMM for one top-k slot.
// Slot lists partition tokens -> each y row written by exactly one block: no atomics.
__global__ __launch_bounds__(256) void moe_sparse_gemm(
    const float* __restrict__ x, const float* __restrict__ Wsp,
    const int* __restrict__ list_idx, const float* __restrict__ list_w,
    const int* __restrict__ counts, float* __restrict__ y, int slot)
{
    const int e = blockIdx.z;
    const int cnt = counts[e * 2 + slot];
    const int m0 = blockIdx.x * MT;
    if (m0 >= cnt) return;                       // uniform across block

    __shared__ float As[MT][36];
    __shared__ float Bs[KT / 2][NT + 2][2];
    __shared__ int   sTok[MT];
    __shared__ float sGate[MT];

    const int tid = threadIdx.x;
    const int lane = tid & 31, w = tid >> 5;
    const int wm = w >> 1, wn = w & 1;
    const int hi = lane >> 4, ml = lane & 15;
    const int n0 = blockIdx.y * NT;
    const int lbase = (e * 2 + slot) * T_TOKENS;

    if (tid < MT) {
        const int p = m0 + tid;
        sTok[tid]  = (p < cnt) ? list_idx[lbase + p] : -1;
        sGate[tid] = (p < cnt) ? list_w [lbase + p] : 0.f;
    }
    __syncthreads();

    const int ar = tid >> 1, ac0 = (tid & 1) << 4;
    const int bn = tid & 127, bh = (tid >> 7) << 3;
    const int  tokR = sTok[ar];
    const float gR  = sGate[ar];
    const float* Wb = Wsp + (size_t)e * DK * DN;

    v8f acc[2][4] = {};

    for (int kk = 0; kk < DK; kk += KT) {
        const float* xa = x + (size_t)tokR * DK + kk + ac0;
#pragma unroll
        for (int j = 0; j < 4; ++j) {
            float4 av = make_float4(0.f, 0.f, 0.f, 0.f);
            if (tokR >= 0) av = *(const float4*)(xa + 4 * j);
            av.x *= gR; av.y *= gR; av.z *= gR; av.w *= gR;   // gate folded into A
            *(float4*)&As[ar][ac0 + 4 * j] = av;
        }
#pragma unroll
        for (int i = 0; i < 8; ++i) {
            const int p = bh + i;
            const float b0 = Wb[(size_t)(kk + 2 * p) * DN + n0 + bn];
            const float b1 = Wb[(size_t)(kk + 2 * p + 1) * DN + n0 + bn];
            const v2f pv = { b0, b1 };
            *(v2f*)&Bs[p][bn][0] = pv;
        }
        if (kk + KT < DK)
            __builtin_prefetch(Wb + (size_t)(kk + KT + 2 * bh) * DN + n0 + bn, 0, 0);
        __syncthreads();

#pragma unroll
        for (int ks = 0; ks < KT; ks += 4) {
            const int kb = ks + (hi << 1);
            const int kp = kb >> 1;
            const v2f a0 = *(const v2f*)&As[wm * 32 + ml][kb];
            const v2f a1 = *(const v2f*)&As[wm * 32 + 16 + ml][kb];
            const v2f b0 = *(const v2f*)&Bs[kp][wn * 64 + ml][0];
            const v2f b1 = *(const v2f*)&Bs[kp][wn * 64 + 16 + ml][0];
            const v2f b2 = *(const v2f*)&Bs[kp][wn * 64 + 32 + ml][0];
            const v2f b3 = *(const v2f*)&Bs[kp][wn * 64 + 48 + ml][0];
            acc[0][0] = __builtin_amdgcn_wmma_f32_16x16x4_f32(false, a0, false, b0, (short)0, acc[0][0], false, false);
            acc[0][1] = __builtin_amdgcn_wmma_f32_16x16x4_f32(false, a0, false, b1, (short)0, acc[0][1], false, false);
            acc[0][2] = __builtin_amdgcn_wmma_f32_16x16x4_f32(false, a0, false, b2, (short)0, acc[0][2], false, false);
            acc[0][3] = __builtin_amdgcn_wmma_f32_16x16x4_f32(false, a0, false, b3, (short)0, acc[0][3], false, false);
            acc[1][0] = __builtin_amdgcn_wmma_f32_16x16x4_f32(false, a1, false, b0, (short)0, acc[1][0], false, false);
            acc[1][1] = __builtin_amdgcn_wmma_f32_16x16x4_f32(false, a1, false, b1, (short)0, acc[1][1], false, false);
            acc[1][2] = __builtin_amdgcn_wmma_f32_16x16x4_f32(false, a1, false, b2, (short)0, acc[1][2], false, false);
            acc[1][3] = __builtin_amdgcn_wmma_f32_16x16x4_f32(false, a1, false, b3, (short)0, acc[1][3], false, false);
        }
        __syncthreads();
    }

#pragma unroll
    for (int sm = 0; sm < 2; ++sm)
#pragma unroll
        for (int v = 0; v < 8; ++v) {
            const int t = sTok[wm * 32 + sm * 16 + v + (hi << 3)];
            if (t >= 0) {                           // row owned exclusively this pass
#pragma unroll
                for (int sn = 0; sn < 4; ++sn) {
                    const int o = n0 + wn * 64 + sn * 16 + ml;
                    y[(size_t)t * DN + o] += acc[sm][sn][v];
                }
            }
        }
}

// ======================================================================
extern "C" void kernel_launch(void* const* d_in, const int* in_sizes, int n_in,
                              void* d_out, int out_size, void* d_ws, size_t ws_size,
                              hipStream_t stream) {
    const float* x   = (const float*)d_in[0];
    const float* Wsh = (const float*)d_in[1];   // [1, DK, DN]
    const float* bsh = (const float*)d_in[2];   // [1, DN]
    const float* Wr  = (const float*)d_in[3];   // [DK, 8]
    const float* br  = (const float*)d_in[4];   // [8]
    const float* Wsp = (const float*)d_in[5];   // [8, DK, DN]
    const float* bsp = (const float*)d_in[6];   // [8, DN]
    float* y = (float*)d_out;

    // workspace carve-out (~1.2 MB)
    char* p = (char*)d_ws;
    int*   top_idx  = (int*)p;   p += (size_t)T_TOKENS * 2 * sizeof(int);
    float* top_w    = (float*)p; p += (size_t)T_TOKENS * 2 * sizeof(float);
    int*   counts   = (int*)p;   p += 64 * sizeof(int);
    int*   list_idx = (int*)p;   p += (size_t)16 * T_TOKENS * sizeof(int);
    float* list_w   = (float*)p; p += (size_t)16 * T_TOKENS * sizeof(float);

    moe_router<<<T_TOKENS / 8, 256, 0, stream>>>(x, Wr, br, top_idx, top_w);
    moe_build_lists<<<16, 256, 0, stream>>>(top_idx, top_w, list_idx, list_w, counts);
    moe_shared_gemm<<<dim3(T_TOKENS / MT, DN / NT), 256, 0, stream>>>(
        x, Wsh, bsh, bsp, top_idx, top_w, y);
    moe_sparse_gemm<<<dim3(T_TOKENS / MT, DN / NT, NE), 256, 0, stream>>>(
        x, Wsp, list_idx, list_w, counts, y, 0);
    moe_sparse_gemm<<<dim3(T_TOKENS / MT, DN / NT, NE), 256, 0, stream>>>(
        x, Wsp, list_idx, list_w, counts, y, 1);
}